// LinearAttention_80161269613187
// MI455X (gfx1250) — compile-verified
//
#include <hip/hip_runtime.h>
#include <hip/hip_bf16.h>

// Linear attention  out = Q @ (K^T @ V),  B=4 H=16 T=4096 D=128, fp32.
// Memory-bound (~536 MB @ 23.3 TB/s => ~23us floor). fp32 WMMA
// (V_WMMA_F32_16X16X4_F32) keeps reference precision; data movement uses
// gfx1250 async-to-LDS (ASYNCcnt) for phase 1 and the Tensor Data Mover
// (TENSORcnt) for phase 2 kv staging.

typedef __attribute__((ext_vector_type(2))) float v2f;
typedef __attribute__((ext_vector_type(8))) float v8f;
typedef __attribute__((ext_vector_type(4))) unsigned int u32x4;
typedef __attribute__((ext_vector_type(4))) int i32x4;
typedef __attribute__((ext_vector_type(8))) int i32x8;

#define T_DIM 4096
#define D_DIM 128
#define TSEG 4                 // T split for phase-1 occupancy (256 blocks)
#define SEG_ROWS (T_DIM / TSEG)
#define CHUNK 32               // t-rows staged per LDS buffer
#define KV_ELEMS (D_DIM * D_DIM)

// ---------------------------------------------------------------------------
// Phase 1: partial[seg][bh][d][e] = sum_{t in seg} K[bh][t][d] * V[bh][t][e]
// 256 blocks = 4 T-segments x 64 heads, 8 waves each; wave w owns d-strip
// [16w,16w+16) over the full 128-wide e range (8 accumulators).
// K/V chunks (32x128 fp32 each) are staged into double-buffered LDS with
// GLOBAL_LOAD_ASYNC_TO_LDS_B128 and consumed by WMMA from LDS.
// ---------------------------------------------------------------------------
__global__ __launch_bounds__(256) void la_kv_kernel(const float* __restrict__ K,
                                                    const float* __restrict__ V,
                                                    float* __restrict__ ws) {
  __shared__ float sK[2][CHUNK * D_DIM];   // 2 x 16 KB
  __shared__ float sV[2][CHUNK * D_DIM];   // 2 x 16 KB  (total 64 KB)

  const int bh   = blockIdx.x & 63;
  const int seg  = blockIdx.x >> 6;
  const int tid  = threadIdx.x;
  const int wave = tid >> 5;
  const int lane = tid & 31;
  const int m    = lane & 15;
  const int half = lane >> 4;            // 0: K-dim {0,1}, 1: K-dim {2,3}
  const int d0   = wave * 16;

  const float* __restrict__ Kbh = K + (size_t)bh * T_DIM * D_DIM;
  const float* __restrict__ Vbh = V + (size_t)bh * T_DIM * D_DIM;
  const int tseg0 = seg * SEG_ROWS;

  const unsigned int ldsK0 = (unsigned int)(uintptr_t)&sK[0][0];
  const unsigned int ldsV0 = (unsigned int)(uintptr_t)&sV[0][0];

  // Each thread async-copies 4 b128 slices of K and 4 of V per chunk
  // (CHUNK*D_DIM/4 = 1024 float4 per matrix / 256 threads).
  auto issue_chunk = [&](int buf, int tbase) {
#pragma unroll
    for (int k2 = 0; k2 < 4; ++k2) {
      const int f   = tid + k2 * 256;          // float4 index in [32 x 32]
      const int row = f >> 5;
      const int c4  = f & 31;
      const float* gK = Kbh + (size_t)(tbase + row) * D_DIM + c4 * 4;
      const float* gV = Vbh + (size_t)(tbase + row) * D_DIM + c4 * 4;
      const unsigned int lK = ldsK0 + (unsigned int)(buf * CHUNK * D_DIM + f * 4) * 4u;
      const unsigned int lV = ldsV0 + (unsigned int)(buf * CHUNK * D_DIM + f * 4) * 4u;
      asm volatile("global_load_async_to_lds_b128 %0, %1, off"
                   :: "v"(lK), "v"(gK) : "memory");
      asm volatile("global_load_async_to_lds_b128 %0, %1, off"
                   :: "v"(lV), "v"(gV) : "memory");
    }
  };

  v8f acc[8] = {};

  issue_chunk(0, tseg0);
  const int nchunks = SEG_ROWS / CHUNK;    // 32
  for (int c = 0; c < nchunks; ++c) {
    asm volatile("s_wait_asynccnt 0x0" ::: "memory");  // own slice landed
    __syncthreads();                                   // => all slices landed
    if (c + 1 < nchunks) issue_chunk((c + 1) & 1, tseg0 + (c + 1) * CHUNK);

    const float* bK = &sK[c & 1][0];
    const float* bV = &sV[c & 1][0];
#pragma unroll
    for (int tt = 0; tt < CHUNK; tt += 4) {
      v2f a;   // A[m][k] = K[t+k][d0+m]   (A = K^T chunk, 16x4)
      a.x = bK[(tt + 2 * half)     * D_DIM + d0 + m];
      a.y = bK[(tt + 2 * half + 1) * D_DIM + d0 + m];
#pragma unroll
      for (int j = 0; j < 8; ++j) {
        v2f b; // B[k][n] = V[t+k][16j+n]  (4x16)
        b.x = bV[(tt + 2 * half)     * D_DIM + j * 16 + m];
        b.y = bV[(tt + 2 * half + 1) * D_DIM + j * 16 + m];
        acc[j] = __builtin_amdgcn_wmma_f32_16x16x4_f32(
            false, a, false, b, (short)0, acc[j], false, false);
      }
    }
  }

  float* __restrict__ part = ws + ((size_t)seg * 64 + bh) * KV_ELEMS;
#pragma unroll
  for (int j = 0; j < 8; ++j) {
#pragma unroll
    for (int i = 0; i < 8; ++i) {
      const int row = d0 + i + 8 * half;   // C/D layout: lanes 16-31 hold M=8+i
      const int col = j * 16 + m;
      part[row * D_DIM + col] = acc[j][i];
    }
  }
}

// ---------------------------------------------------------------------------
// Reduce: kv[bh] (= ws[0][bh]) += partials from segments 1..3. Deterministic
// tree-free per-element sum (no float atomics). Tiny: 16 MB traffic.
// ---------------------------------------------------------------------------
__global__ __launch_bounds__(256) void la_reduce_kernel(float* __restrict__ ws) {
  const int bh = blockIdx.x;
  float4* base = (float4*)ws + (size_t)bh * (KV_ELEMS / 4);
  const size_t segstride = (size_t)64 * (KV_ELEMS / 4);
#pragma unroll
  for (int i = 0; i < 16; ++i) {
    const int idx = threadIdx.x + i * 256;   // 4096 float4 per head
    float4 a  = base[idx];
    float4 b1 = base[idx + segstride];
    float4 b2 = base[idx + 2 * segstride];
    float4 b3 = base[idx + 3 * segstride];
    a.x += b1.x + b2.x + b3.x;
    a.y += b1.y + b2.y + b3.y;
    a.z += b1.z + b2.z + b3.z;
    a.w += b1.w + b2.w + b3.w;
    base[idx] = a;
  }
}

// ---------------------------------------------------------------------------
// Phase 2: out[bh][t][e] = sum_d Q[bh][t][d] * kv[bh][d][e]
// 2048 blocks (64 heads x 32 t-chunks of 128 rows). The 64 KB kv tile is
// staged into LDS by the Tensor Data Mover (one tensor_load_to_lds per
// block, wave 0), then 8 waves each compute a 16(t) x 128(e) strip.
// ---------------------------------------------------------------------------
__global__ __launch_bounds__(256) void la_out_kernel(const float* __restrict__ Q,
                                                     const float* __restrict__ kv,
                                                     float* __restrict__ Out) {
  __shared__ float skv[KV_ELEMS];   // 64 KB

  const int bh    = blockIdx.x >> 5;
  const int chunk = blockIdx.x & 31;
  const float* __restrict__ kvbh = kv + (size_t)bh * KV_ELEMS;

#if __has_builtin(__builtin_amdgcn_tensor_load_to_lds)
  if (threadIdx.x < 32) {     // one TDM issue per block (wave-level op)
    const unsigned long long ga = (unsigned long long)(uintptr_t)kvbh;
    const unsigned int lds     = (unsigned int)(uintptr_t)&skv[0];
    // D# group 0: count=1 | lds_addr | global_addr(57b) | type=2
    u32x4 g0;
    g0[0] = 1u;
    g0[1] = lds;
    g0[2] = (unsigned int)ga;
    g0[3] = (unsigned int)(ga >> 32) | (2u << 30);
    // D# group 1: data_size=4B; tensor 128x128, tile 128x128, stride 128
    i32x8 g1;
    g1[0] = 0x00020000;      // workgroup_mask=0, data_size=2 (4 bytes)
    g1[1] = 128 << 16;       // tensor_dim0[15:0]
    g1[2] = 128 << 16;       // tensor_dim0[31:16]=0 | tensor_dim1[15:0]
    g1[3] = 128 << 16;       // tensor_dim1[31:16]=0 | tile_dim0=128
    g1[4] = 128;             // tile_dim1=128 | tile_dim2=0
    g1[5] = 128;             // tensor_dim0_stride[31:0]
    g1[6] = 0;
    g1[7] = 0;
    const i32x4 z4 = {};
#if __clang_major__ >= 23
    const i32x8 z8 = {};
    __builtin_amdgcn_tensor_load_to_lds(g0, g1, z4, z4, z8, 0);
#else
    __builtin_amdgcn_tensor_load_to_lds(g0, g1, z4, z4, 0);
#endif
    __builtin_amdgcn_s_wait_tensorcnt((short)0);
  }
  __syncthreads();
#else
  // Fallback: cooperative b128 staging.
  {
    const float4* __restrict__ src = (const float4*)kvbh;
    float4* dst = (float4*)skv;
#pragma unroll
    for (int i = 0; i < 16; ++i) {
      const int idx = threadIdx.x + i * 256;
      dst[idx] = src[idx];
    }
  }
  __syncthreads();
#endif

  const int wave = threadIdx.x >> 5;
  const int lane = threadIdx.x & 31;
  const int m    = lane & 15;
  const int half = lane >> 4;
  const int t0   = chunk * 128 + wave * 16;

  const float* __restrict__ Qbh = Q + (size_t)bh * T_DIM * D_DIM;
  // A[m][k] = Q[t0+m][d0+k]; per lane, (a.x,a.y) is one aligned b64 load.
  const float* qrow = Qbh + (size_t)(t0 + m) * D_DIM + 2 * half;

  v8f acc[8] = {};

  for (int dd = 0; dd < D_DIM; dd += 4) {
    const v2f a = *(const v2f*)(qrow + dd);
#pragma unroll
    for (int j = 0; j < 8; ++j) {
      v2f b;
      b.x = skv[(dd + 2 * half)     * D_DIM + j * 16 + m];  // B[0][n]/B[2][n]
      b.y = skv[(dd + 2 * half + 1) * D_DIM + j * 16 + m];  // B[1][n]/B[3][n]
      acc[j] = __builtin_amdgcn_wmma_f32_16x16x4_f32(
          false, a, false, b, (short)0, acc[j], false, false);
    }
  }

  float* __restrict__ Outbh = Out + (size_t)bh * T_DIM * D_DIM;
#pragma unroll
  for (int j = 0; j < 8; ++j) {
#pragma unroll
    for (int i = 0; i < 8; ++i) {
      const int row = t0 + i + 8 * half;
      const int col = j * 16 + m;
      Outbh[(size_t)row * D_DIM + col] = acc[j][i];
    }
  }
}

// ---------------------------------------------------------------------------
extern "C" void kernel_launch(void* const* d_in, const int* in_sizes, int n_in,
                              void* d_out, int out_size, void* d_ws, size_t ws_size,
                              hipStream_t stream) {
  const float* Q = (const float*)d_in[0];
  const float* K = (const float*)d_in[1];
  const float* V = (const float*)d_in[2];
  float* out = (float*)d_out;
  float* ws  = (float*)d_ws;   // 4 segs x 64 heads x 128x128 fp32 = 16 MB

  // Phase 1: 4 T-segments x 64 heads.
  la_kv_kernel<<<dim3(TSEG * 64), dim3(256), 0, stream>>>(K, V, ws);
  // Segment reduction into ws[0][bh] (deterministic, no atomics).
  la_reduce_kernel<<<dim3(64), dim3(256), 0, stream>>>(ws);
  // Phase 2: 64 heads x 32 t-chunks.
  la_out_kernel<<<dim3(64 * 32), dim3(256), 0, stream>>>(Q, ws, out);
}